// GatedGCNNet_50440095924267
// MI455X (gfx1250) — compile-verified
//
#include <hip/hip_runtime.h>
#include <hip/hip_bf16.h>

#define DIMC 128
#define NNODES 20000
#define NEDGES 256000
#define NLAYERS 4

typedef float v2f __attribute__((ext_vector_type(2)));
typedef float v4f __attribute__((ext_vector_type(4)));
typedef float v8f __attribute__((ext_vector_type(8)));

// ---------------------------------------------------------------------------
// C[M,128] = A[M,128] @ W[128,128] + bias[128]   (fp32 WMMA 16x16x4)
// Block: 256 threads = 8 waves. Each wave computes one 16-row x 128-col panel.
// W staged in LDS in K-pair-interleaved form so every B fragment is ONE
// naturally even-aligned ds_load_b64:
//   Wl[(k>>1)*256 + n*2 + (k&1)] = W[k][n]
// Fragment for k-step k0, lane-half hl, column n:
//   { W[k0+2*hl][n], W[k0+2*hl+1][n] } = *(v2f*)&Wl[((k0>>1)+hl)*256 + n*2]
// M must be a multiple of 16 (20000 and 256000 are).
// ---------------------------------------------------------------------------
__global__ __launch_bounds__(256) void gemm_bias(
    const float* __restrict__ A, const float* __restrict__ W,
    const float* __restrict__ bias, float* __restrict__ C, int M) {
  __shared__ __align__(16) float Wl[DIMC * DIMC];
  __shared__ float bl[DIMC];
  const int tid = threadIdx.x;

  // cooperative load of W (16384 floats = 4096 float4) into interleaved layout
  const v4f* W4 = (const v4f*)W;
#pragma unroll
  for (int i = 0; i < 16; ++i) {
    const int idx = tid + i * 256;       // float4 index
    const v4f w = W4[idx];
    const int k = idx >> 5;              // row (idx*4 >> 7)
    const int n = (idx & 31) * 4;        // starting column
    float* dstp = Wl + (k >> 1) * 256 + (k & 1);
#pragma unroll
    for (int j = 0; j < 4; ++j) dstp[(n + j) * 2] = w[j];
  }
  if (tid < DIMC) bl[tid] = bias[tid];
  __syncthreads();

  const int wave = tid >> 5;
  const int lane = tid & 31;
  const int tile = blockIdx.x * 8 + wave;
  if (tile * 16 >= M) return;  // wave-uniform guard: EXEC all-1s inside

  const int hl  = lane >> 4;   // 0: lanes 0-15 (K=0,1), 1: lanes 16-31 (K=2,3)
  const int l15 = lane & 15;
  const float* Arow = A + (size_t)(tile * 16 + l15) * DIMC;

  v8f acc[8];
#pragma unroll
  for (int nt = 0; nt < 8; ++nt) {
    const float bv = bl[nt * 16 + l15];  // column = nt*16 + (lane&15)
#pragma unroll
    for (int r = 0; r < 8; ++r) acc[nt][r] = bv;
  }

  const float* Wlane = Wl + hl * 256 + l15 * 2;  // per-lane base into LDS
#pragma unroll 4
  for (int k0 = 0; k0 < DIMC; k0 += 4) {
    // A fragment: contiguous 8B -> global_load_b64
    const v2f a = *(const v2f*)(Arow + k0 + hl * 2);
    const float* Wrow = Wlane + (k0 >> 1) * 256;
#pragma unroll
    for (int nt = 0; nt < 8; ++nt) {
      const v2f b = *(const v2f*)(Wrow + nt * 32);  // one ds_load_b64
      acc[nt] = __builtin_amdgcn_wmma_f32_16x16x4_f32(
          false, a, false, b, (short)0, acc[nt], false, false);
    }
  }

  // C/D layout: VGPR r, lanes 0-15 -> M=r, lanes 16-31 -> M=r+8; N = nt*16+l15
#pragma unroll
  for (int nt = 0; nt < 8; ++nt) {
#pragma unroll
    for (int r = 0; r < 8; ++r) {
      C[(size_t)(tile * 16 + r + hl * 8) * DIMC + nt * 16 + l15] = acc[nt][r];
    }
  }
}

// ---------------------------------------------------------------------------
// Per-edge: e_new = Dh[src] + Eh[dst] + Ce; sigma = sigmoid(e_new);
// num[dst] += sigma * Bh[src]; den[dst] += sigma.  One wave per edge.
// ---------------------------------------------------------------------------
__global__ __launch_bounds__(256) void edge_gate(
    const float* __restrict__ Dh, const float* __restrict__ Eh,
    const float* __restrict__ Bh, const int* __restrict__ src,
    const int* __restrict__ dst, float* __restrict__ enew,
    float* num, float* den) {
  const int eid = blockIdx.x * 8 + (threadIdx.x >> 5);
  if (eid >= NEDGES) return;
  const int lane = threadIdx.x & 31;
  const int c = lane * 4;
  const int s = src[eid];
  const int d = dst[eid];

  v4f dh = *(const v4f*)(Dh + (size_t)s * DIMC + c);
  v4f eh = *(const v4f*)(Eh + (size_t)d * DIMC + c);
  v4f ce = *(const v4f*)(enew + (size_t)eid * DIMC + c);
  v4f en = dh + eh + ce;
  *(v4f*)(enew + (size_t)eid * DIMC + c) = en;

  v4f bh = *(const v4f*)(Bh + (size_t)s * DIMC + c);
  float* np = num + (size_t)d * DIMC + c;
  float* dp = den + (size_t)d * DIMC + c;
#pragma unroll
  for (int i = 0; i < 4; ++i) {
    const float sg = 1.0f / (1.0f + __expf(-en[i]));
    atomicAdd(np + i, sg * bh[i]);   // global_atomic_add_f32
    atomicAdd(dp + i, sg);
  }
}

// ---------------------------------------------------------------------------
// Column stats (sum, sumsq) over R rows of X[R,128] -> stats[0:128]=sum,
// stats[128:256]=sumsq. Coalesced reads, LDS reduce, per-block atomics.
// ---------------------------------------------------------------------------
__global__ __launch_bounds__(256) void col_stats(
    const float* __restrict__ X, int R, float* stats) {
  __shared__ float s1[256];
  __shared__ float s2[256];
  const int tid = threadIdx.x;
  const int col = tid & 127;
  const int hf = tid >> 7;
  float sum = 0.f, sq = 0.f;
  for (int r = blockIdx.x * 2 + hf; r < R; r += gridDim.x * 2) {
    const float x = X[(size_t)r * DIMC + col];
    sum += x;
    sq += x * x;
  }
  s1[tid] = sum;
  s2[tid] = sq;
  __syncthreads();
  if (tid < 128) {
    atomicAdd(&stats[tid], s1[tid] + s1[tid + 128]);
    atomicAdd(&stats[128 + tid], s2[tid] + s2[tid + 128]);
  }
}

// h_new = Ah + num / (den + 1e-6)
__global__ void node_combine(const float* __restrict__ Ah,
                             const float* __restrict__ num,
                             const float* __restrict__ den,
                             float* __restrict__ hnew) {
  const size_t i = (size_t)blockIdx.x * blockDim.x + threadIdx.x;
  if (i < (size_t)NNODES * DIMC) hnew[i] = Ah[i] + num[i] / (den[i] + 1e-6f);
}

// out = res + relu(gamma*(x-mu)*rsqrt(var+eps) + beta)   (res/out may alias)
__global__ void bn_relu_res(const float* __restrict__ X,
                            const float* __restrict__ stats,
                            const float* __restrict__ gamma,
                            const float* __restrict__ beta,
                            const float* res, float* out, int R) {
  const size_t i = (size_t)blockIdx.x * blockDim.x + threadIdx.x;
  if (i >= (size_t)R * DIMC) return;
  const int col = (int)(i & (DIMC - 1));
  const float invR = 1.0f / (float)R;
  const float mu = stats[col] * invR;
  const float var = stats[128 + col] * invR - mu * mu;
  const float xn = gamma[col] * (X[i] - mu) * rsqrtf(var + 1e-5f) + beta[col];
  out[i] = res[i] + fmaxf(xn, 0.0f);
}

// ---------------------------------------------------------------------------
extern "C" void kernel_launch(void* const* d_in, const int* in_sizes, int n_in,
                              void* d_out, int out_size, void* d_ws,
                              size_t ws_size, hipStream_t stream) {
  (void)in_sizes; (void)n_in; (void)out_size; (void)ws_size;

  const float* h0      = (const float*)d_in[0];
  const float* e0      = (const float*)d_in[1];
  const int*   src     = (const int*)d_in[2];
  const int*   dst     = (const int*)d_in[3];
  const float* W_emb_h = (const float*)d_in[4];
  const float* b_emb_h = (const float*)d_in[5];
  const float* W_emb_e = (const float*)d_in[6];
  const float* b_emb_e = (const float*)d_in[7];
  const float* Aw = (const float*)d_in[8];
  const float* Ab = (const float*)d_in[9];
  const float* Bw = (const float*)d_in[10];
  const float* Bb = (const float*)d_in[11];
  const float* Cw = (const float*)d_in[12];
  const float* Cb = (const float*)d_in[13];
  const float* Dw = (const float*)d_in[14];
  const float* Db = (const float*)d_in[15];
  const float* Ew = (const float*)d_in[16];
  const float* Eb = (const float*)d_in[17];
  const float* gh = (const float*)d_in[18];
  const float* bh = (const float*)d_in[19];
  const float* ge = (const float*)d_in[20];
  const float* be = (const float*)d_in[21];

  const size_t NH = (size_t)NNODES * DIMC;  // 2,560,000 floats
  const size_t NE = (size_t)NEDGES * DIMC;  // 32,768,000 floats

  float* ws = (float*)d_ws;
  float* h    = ws; ws += NH;
  float* e    = ws; ws += NE;
  float* Ahb  = ws; ws += NH;
  float* Bhb  = ws; ws += NH;
  float* Dhb  = ws; ws += NH;
  float* Ehb  = ws; ws += NH;
  float* enew = ws; ws += NE;
  float* num  = ws; ws += NH;
  float* den  = ws; ws += NH;
  float* hnew = ws; ws += NH;
  float* stats = ws; ws += 256;

  const int grid_gh = (NNODES / 16 + 7) / 8;  // 157 blocks (1250 tiles)
  const int grid_ge = (NEDGES / 16) / 8;      // 2000 blocks (16000 tiles)
  const int grid_nh = (int)((NH + 255) / 256);
  const int grid_ne = (int)((NE + 255) / 256);

  // input embeddings
  gemm_bias<<<grid_gh, 256, 0, stream>>>(h0, W_emb_h, b_emb_h, h, NNODES);
  gemm_bias<<<grid_ge, 256, 0, stream>>>(e0, W_emb_e, b_emb_e, e, NEDGES);

  for (int l = 0; l < NLAYERS; ++l) {
    const float* Awl = Aw + (size_t)l * DIMC * DIMC;
    const float* Bwl = Bw + (size_t)l * DIMC * DIMC;
    const float* Cwl = Cw + (size_t)l * DIMC * DIMC;
    const float* Dwl = Dw + (size_t)l * DIMC * DIMC;
    const float* Ewl = Ew + (size_t)l * DIMC * DIMC;
    const float* Abl = Ab + (size_t)l * DIMC;
    const float* Bbl = Bb + (size_t)l * DIMC;
    const float* Cbl = Cb + (size_t)l * DIMC;
    const float* Dbl = Db + (size_t)l * DIMC;
    const float* Ebl = Eb + (size_t)l * DIMC;

    gemm_bias<<<grid_gh, 256, 0, stream>>>(h, Awl, Abl, Ahb, NNODES);
    gemm_bias<<<grid_gh, 256, 0, stream>>>(h, Bwl, Bbl, Bhb, NNODES);
    gemm_bias<<<grid_gh, 256, 0, stream>>>(h, Dwl, Dbl, Dhb, NNODES);
    gemm_bias<<<grid_gh, 256, 0, stream>>>(h, Ewl, Ebl, Ehb, NNODES);
    gemm_bias<<<grid_ge, 256, 0, stream>>>(e, Cwl, Cbl, enew, NEDGES);

    hipMemsetAsync(num, 0, NH * sizeof(float), stream);
    hipMemsetAsync(den, 0, NH * sizeof(float), stream);

    edge_gate<<<NEDGES / 8, 256, 0, stream>>>(Dhb, Ehb, Bhb, src, dst, enew,
                                              num, den);

    node_combine<<<grid_nh, 256, 0, stream>>>(Ahb, num, den, hnew);

    // BatchNorm(h_new) + relu + residual
    hipMemsetAsync(stats, 0, 256 * sizeof(float), stream);
    col_stats<<<256, 256, 0, stream>>>(hnew, NNODES, stats);
    float* hout = (l == NLAYERS - 1) ? (float*)d_out : h;
    bn_relu_res<<<grid_nh, 256, 0, stream>>>(hnew, stats, gh + l * DIMC,
                                             bh + l * DIMC, h, hout, NNODES);

    // BatchNorm(e_new) + relu + residual (skip on last layer: e unused)
    if (l < NLAYERS - 1) {
      hipMemsetAsync(stats, 0, 256 * sizeof(float), stream);
      col_stats<<<256, 256, 0, stream>>>(enew, NEDGES, stats);
      bn_relu_res<<<grid_ne, 256, 0, stream>>>(enew, stats, ge + l * DIMC,
                                               be + l * DIMC, e, e, NEDGES);
    }
  }
}